// PixelwiseNet_28698971472073
// MI455X (gfx1250) — compile-verified
//
#include <hip/hip_runtime.h>

typedef __attribute__((ext_vector_type(2))) float v2f;
typedef __attribute__((ext_vector_type(8))) float v8f;

#define BN_EPS 1e-5f

// Problem dims (fixed by the reference harness)
constexpr int  Bn  = 4, Cin = 8, Dd = 48, Hh = 128, Ww = 160;
constexpr int  HW  = Hh * Ww;                 // 20480 pixels per (b,d) slice
constexpr long CHS = (long)Dd * HW;           // channel stride = 983040 elems
constexpr int  TILES_PER_B = HW / 16;         // 1280 16-pixel tiles per batch
constexpr int  NTILES      = Bn * TILES_PER_B;// 5120 output tiles
constexpr int  DSPLIT      = 4;               // waves per output tile (depth split)
constexpr int  DPER        = Dd / DSPLIT;     // 12 depths per wave
constexpr int  NJOBS       = NTILES * DSPLIT; // 20480 wave jobs
constexpr int  WAVES_PER_BLOCK = 8;           // 256 threads (wave32)
constexpr int  NBLOCKS     = NJOBS / WAVES_PER_BLOCK; // 2560

__global__ void init_out_kernel(unsigned int* __restrict__ out, int n) {
    int i = blockIdx.x * blockDim.x + threadIdx.x;
    if (i < n) out[i] = 0u;   // bits of +0.0f; sigmoid > 0 so max is safe
}

__global__ __launch_bounds__(256) void pixelnet_wmma_kernel(
    const float* __restrict__ x1,
    const float* __restrict__ w0, const float* __restrict__ g0,
    const float* __restrict__ b0, const float* __restrict__ m0, const float* __restrict__ v0,
    const float* __restrict__ w1, const float* __restrict__ g1,
    const float* __restrict__ b1, const float* __restrict__ m1, const float* __restrict__ v1,
    const float* __restrict__ w2, const float* __restrict__ bias2,
    unsigned int* __restrict__ out)
{
    const int lane = threadIdx.x & 31;
    const int wave = threadIdx.x >> 5;
    const int job  = blockIdx.x * WAVES_PER_BLOCK + wave;   // exactly NJOBS jobs
    const int tile = job >> 2;           // / DSPLIT
    const int z    = job & 3;            // depth slice of this wave
    const int b    = tile / TILES_PER_B;
    const int s0   = (tile % TILES_PER_B) * 16;             // pixel base in [0,HW)

    const bool lo   = (lane < 16);
    const int  mrow = lane & 15;
    const int  kb   = lo ? 0 : 2;        // K pair held by this half-wave (A/B layout)

    // ---------------- fold BN into weights; build per-lane WMMA operands ---------
    // Layer 1: A1 = diag(scale0) * W0  (16x8), split into two 16x4 operands.
    const float sc0 = g0[mrow] * rsqrtf(v0[mrow] + BN_EPS);
    v2f a1lo, a1hi;
    a1lo.x = w0[mrow * 8 + kb + 0] * sc0;     // K = kb
    a1lo.y = w0[mrow * 8 + kb + 1] * sc0;     // K = kb+1
    a1hi.x = w0[mrow * 8 + kb + 4] * sc0;     // K = 4+kb
    a1hi.y = w0[mrow * 8 + kb + 5] * sc0;     // K = 5+kb
    v8f c1;                                    // BN shift as C-matrix bias
    #pragma unroll
    for (int r = 0; r < 8; ++r) {
        const int M = r + (lo ? 0 : 8);
        const float s = g0[M] * rsqrtf(v0[M] + BN_EPS);
        c1[r] = b0[M] - m0[M] * s;
    }
    // Layer 2: A2 = diag(scale1) * W1 (8x16), zero-padded to 16x16, four 16x4 ops.
    const float sc1 = (mrow < 8) ? g1[mrow] * rsqrtf(v1[mrow] + BN_EPS) : 0.0f;
    v2f a2[4];
    #pragma unroll
    for (int kk = 0; kk < 4; ++kk) {
        const int k2 = 4 * kk + kb;
        a2[kk].x = (mrow < 8) ? w1[mrow * 16 + k2 + 0] * sc1 : 0.0f;
        a2[kk].y = (mrow < 8) ? w1[mrow * 16 + k2 + 1] * sc1 : 0.0f;
    }
    v8f c2;
    #pragma unroll
    for (int r = 0; r < 8; ++r) {
        const float s = g1[r] * rsqrtf(v1[r] + BN_EPS);
        c2[r] = lo ? (b1[r] - m1[r] * s) : 0.0f;   // padded rows get zero bias
    }
    float w2s[8];
    #pragma unroll
    for (int j = 0; j < 8; ++j) w2s[j] = w2[j];
    const float bias2v = bias2[0];

    // ---------------- streaming loop over this wave's depth slice ----------------
    const float* xb  = x1 + (long)b * Cin * CHS + s0;
    const long  off0 = (long)kb * CHS + mrow;   // (channel kb, pixel mrow)
    const int   d0   = z * DPER;

    float rmax = 0.0f;
    for (int dd = 0; dd < DPER; ++dd) {
        const float* xd = xb + (long)(d0 + dd) * HW;
        // prefetch next depth slice (global_prefetch_b8)
        const float* xn = xb + (long)(d0 + ((dd + 1 < DPER) ? dd + 1 : dd)) * HW;
        __builtin_prefetch(xn + off0, 0, 0);
        __builtin_prefetch(xn + off0 + 4 * CHS, 0, 0);

        // Load X tile (8 ch x 16 px) directly in WMMA B-layout: 4 coalesced loads.
        v2f bx_lo, bx_hi;
        bx_lo.x = xd[off0];                // K = kb
        bx_lo.y = xd[off0 + CHS];          // K = kb+1
        bx_hi.x = xd[off0 + 4 * CHS];      // K = 4+kb
        bx_hi.y = xd[off0 + 5 * CHS];      // K = 5+kb

        // Layer 1: h(16x16) = W0' * X + shift0  (two K=4 WMMAs), then ReLU.
        v8f h = c1;
        h = __builtin_amdgcn_wmma_f32_16x16x4_f32(false, a1lo, false, bx_lo,
                                                  (short)0, h, false, false);
        h = __builtin_amdgcn_wmma_f32_16x16x4_f32(false, a1hi, false, bx_hi,
                                                  (short)0, h, false, false);
        #pragma unroll
        for (int r = 0; r < 8; ++r) h[r] = fmaxf(h[r], 0.0f);

        // Layer 2: remap h from D-layout to B-layout via wave32 shuffles,
        // then four K=4 WMMAs: y(8x16, padded) = W1' * h + shift1.
        v8f acc2 = c2;
        #pragma unroll
        for (int kk = 0; kk < 4; ++kk) {
            const int srcl = mrow + ((kk >= 2) ? 16 : 0); // ch>=8 live in high lanes
            const int cl   = (4 * kk) & 7;                // h-register of low-half ch
            v2f bb;
            const float x_lo = __shfl(h[cl + 0], srcl, 32);
            const float x_hi = __shfl(h[cl + 2], srcl, 32);
            bb.x = lo ? x_lo : x_hi;
            const float y_lo = __shfl(h[cl + 1], srcl, 32);
            const float y_hi = __shfl(h[cl + 3], srcl, 32);
            bb.y = lo ? y_lo : y_hi;
            acc2 = __builtin_amdgcn_wmma_f32_16x16x4_f32(false, a2[kk], false, bb,
                                                         (short)0, acc2, false, false);
        }

        // Layer 3 (8->1) + sigmoid per pixel (valid in lanes 0..15; high lanes = 0s).
        float zv = bias2v;
        #pragma unroll
        for (int j = 0; j < 8; ++j) zv = fmaf(w2s[j], fmaxf(acc2[j], 0.0f), zv);
        const float sg = 1.0f / (1.0f + __expf(-zv));
        rmax = fmaxf(rmax, sg);
    }

    // Depth-max combine across the DSPLIT waves sharing this tile.
    // All values are positive floats -> u32 bit compare == float compare.
    if (lo) {
        atomicMax(&out[b * HW + s0 + mrow], __float_as_uint(rmax));
    }
}

extern "C" void kernel_launch(void* const* d_in, const int* in_sizes, int n_in,
                              void* d_out, int out_size, void* d_ws, size_t ws_size,
                              hipStream_t stream) {
    const float* x1    = (const float*)d_in[0];
    const float* w0    = (const float*)d_in[1];
    const float* g0    = (const float*)d_in[2];
    const float* b0    = (const float*)d_in[3];
    const float* m0    = (const float*)d_in[4];
    const float* v0    = (const float*)d_in[5];
    const float* w1    = (const float*)d_in[6];
    const float* g1    = (const float*)d_in[7];
    const float* b1    = (const float*)d_in[8];
    const float* m1    = (const float*)d_in[9];
    const float* v1    = (const float*)d_in[10];
    const float* w2    = (const float*)d_in[11];
    const float* bias2 = (const float*)d_in[12];
    unsigned int* out  = (unsigned int*)d_out;

    const int nOut = Bn * HW; // 81920
    init_out_kernel<<<(nOut + 255) / 256, 256, 0, stream>>>(out, nOut);
    pixelnet_wmma_kernel<<<NBLOCKS, 256, 0, stream>>>(
        x1, w0, g0, b0, m0, v0, w1, g1, b1, m1, v1, w2, bias2, out);
}